// SoftmaxCategoricalHead_64896955842631
// MI455X (gfx1250) — compile-verified
//
#include <hip/hip_runtime.h>
#include <math.h>

// ---------------- problem constants (from reference) ----------------
#define B_    16
#define S_    64
#define V_    50257
#define P_    64
#define TEMP_ 0.8f
#define TOPK_ 50
#define TOPP_ 0.9f
#define REP_  1.2f
#define FILT_ (-1e9f)

#define NT_      256                       // 8 wave32 per block
#define SURV_CAP 128                       // top-k survivors (50 + tie slack)
#define VPAD_    (((V_ + 3) & ~3) + 4)     // +4 slack for per-row alignment shift
#define SMEMW_   (VPAD_ + 256 + SURV_CAP + 8)
#define SMEMB_   (SMEMW_ * 4)              // ~202.6 KB of the 320 KB/WGP LDS
#define NEGPAD_  (-3.0e38f)                // pad value, can never reach top-50

// ---------------- CDNA5 async global->LDS helpers ----------------
typedef int v4i_t __attribute__((ext_vector_type(4)));
typedef __attribute__((address_space(1))) v4i_t* g_v4i_p;   // global v4i*
typedef __attribute__((address_space(3))) v4i_t* l_v4i_p;   // LDS    v4i*
typedef __attribute__((address_space(1))) int*   g_i32_p;   // global int*
typedef __attribute__((address_space(3))) int*   l_i32_p;   // LDS    int*

__device__ __forceinline__ void async_cp_b128(const float* g, float* l) {
#if __has_builtin(__builtin_amdgcn_global_load_async_to_lds_b128)
  __builtin_amdgcn_global_load_async_to_lds_b128(
      (g_v4i_p)(unsigned long long)g,
      // flat LDS address: low 32 bits == LDS byte offset (ISA 10.2 aperture map)
      (l_v4i_p)(unsigned)(unsigned long long)l, 0, 0);
#elif defined(__AMDGCN__)
  unsigned lo = (unsigned)(unsigned long long)l;
  unsigned long long ga = (unsigned long long)g;
  asm volatile("global_load_async_to_lds_b128 %0, %1, off" :: "v"(lo), "v"(ga) : "memory");
#endif
}
__device__ __forceinline__ void async_cp_b32(const float* g, float* l) {
#if __has_builtin(__builtin_amdgcn_global_load_async_to_lds_b32)
  __builtin_amdgcn_global_load_async_to_lds_b32(
      (g_i32_p)(unsigned long long)g,
      (l_i32_p)(unsigned)(unsigned long long)l, 0, 0);
#elif defined(__AMDGCN__)
  unsigned lo = (unsigned)(unsigned long long)l;
  unsigned long long ga = (unsigned long long)g;
  asm volatile("global_load_async_to_lds_b32 %0, %1, off" :: "v"(lo), "v"(ga) : "memory");
#endif
}
__device__ __forceinline__ void wait_async0() {
#if __has_builtin(__builtin_amdgcn_s_wait_asynccnt)
  __builtin_amdgcn_s_wait_asynccnt(0);
#elif defined(__AMDGCN__)
  asm volatile("s_wait_asynccnt 0" ::: "memory");
#endif
}

// monotonic float->uint map: order-preserving for all finite floats
__device__ __forceinline__ unsigned fmap(float f) {
  unsigned u = __float_as_uint(f);
  return (u & 0x80000000u) ? ~u : (u | 0x80000000u);
}

// ---------------- kernel 1: repetition penalty pre-pass ----------------
// For each (s,p): gather logits[b,s,tok] over all b, compute all_neg across the
// batch, store RAW-domain penalized values to workspace [B*S, P]. Temperature
// is applied lazily (division by positive T is order-preserving).
__global__ void __launch_bounds__(NT_)
pen_kernel(const float* __restrict__ logits, const int* __restrict__ prev,
           float* __restrict__ pen) {
  int idx = blockIdx.x * NT_ + threadIdx.x;
  if (idx >= S_ * P_) return;
  int s = idx / P_;
  int p = idx - s * P_;
  int tok = prev[idx];

  float g[B_];
  bool all_neg = true;
#pragma unroll
  for (int b = 0; b < B_; ++b) {
    g[b] = logits[(size_t)(b * S_ + s) * V_ + tok];
    all_neg = all_neg && (g[b] < 0.0f);
  }
#pragma unroll
  for (int b = 0; b < B_; ++b) {
    float pv = all_neg ? g[b] * REP_ : g[b] / REP_;   // match reference mul/div
    pen[(size_t)(b * S_ + s) * P_ + p] = pv;          // raw domain (no /T yet)
  }
}

// ---------------- kernel 2: one workgroup per (b,s) row ----------------
__global__ void __launch_bounds__(NT_)
row_kernel(const float* __restrict__ logits, const int* __restrict__ prev,
           const float* __restrict__ pen, float* __restrict__ out) {
  extern __shared__ __align__(16) unsigned char smem[];
  float*    store = (float*)smem;                    // VPAD_ floats (row + shift slack)
  unsigned* hist  = (unsigned*)(store + VPAD_);      // 256 bins
  float*    surv  = (float*)(hist + 256);            // SURV_CAP survivors
  unsigned* mi    = (unsigned*)(surv + SURV_CAP);    // [0]=prefix [1]=k [2]=count
  float*    mf    = (float*)(mi + 4);                // [0]=T2raw [1]=logZ

  const int tid = threadIdx.x;
  const int r   = blockIdx.x;           // r = b*S + s
  const int s   = r % S_;
  const float* grow = logits + (size_t)r * V_;

  // Per-row alignment: element offset r*V_ == r (mod 4). Shift the LDS image by
  // sh = r&3 elements so that global-16B-aligned chunks are also LDS-16B-aligned.
  const unsigned sh = (unsigned)r & 3u;
  float* row = store + sh;
  const int a0 = (int)((4u - sh) & 3u);          // first 16B-aligned element
  const int nq = (V_ - a0) / 4;                  // # of b128 chunks
  const int t0 = a0 + 4 * nq;                    // tail start (0..3 tail elems)

  // ---- stage the whole raw row HBM -> LDS with the async copy engine ----
  if (tid < a0)        async_cp_b32(grow + tid, row + tid);
  for (int q = tid; q < nq; q += NT_)
                       async_cp_b128(grow + a0 + 4 * q, row + a0 + 4 * q);
  if (tid < V_ - t0)   async_cp_b32(grow + t0 + tid, row + t0 + tid);
  wait_async0();                                  // per-wave ASYNCcnt drain
  __syncthreads();                                // cross-wave LDS visibility

  // ---- pad slots, counters, penalty scatter (raw-domain, dup-safe) ----
  if (tid < (int)sh) store[tid] = NEGPAD_;        // lead pads
  {
    const int hi = (int)sh + V_;                  // tail pads
    if (tid < VPAD_ - hi) store[hi + tid] = NEGPAD_;
  }
  if (tid == 0) { mi[0] = 0u; mi[1] = TOPK_; }
  if (tid < P_) {
    int tok = prev[s * P_ + tid];
    row[tok] = pen[(size_t)r * P_ + tid];
  }
  __syncthreads();

  // ---- 4-pass 8-bit radix select of the 50th-largest (raw bits, exact) ----
  for (int pass = 0; pass < 4; ++pass) {
    hist[tid] = 0u;
    __syncthreads();
    const unsigned prefix = mi[0];
    const unsigned kk     = mi[1];
    const int shift = 24 - 8 * pass;
    for (int j = tid * 4; j < VPAD_; j += NT_ * 4) {
      float4 v = *(const float4*)(store + j);     // aligned b128 LDS load
      float e[4] = {v.x, v.y, v.z, v.w};
#pragma unroll
      for (int t = 0; t < 4; ++t) {
        unsigned u = fmap(e[t]);
        bool ok = (pass == 0) || ((u >> (shift + 8)) == (prefix >> (shift + 8)));
        if (ok) atomicAdd(&hist[(u >> shift) & 0xFFu], 1u);
      }
    }
    __syncthreads();
    if (tid == 0) {
      unsigned c = 0; int bsel = 0;
      for (int bb = 255; bb >= 0; --bb) {
        unsigned h = hist[bb];
        if (c + h >= kk) { bsel = bb; break; }
        c += h;
      }
      mi[0] = prefix | ((unsigned)bsel << shift);
      mi[1] = kk - c;
      if (pass == 3) mi[2] = 0u;                  // survivor counter
    }
    __syncthreads();
  }
  const unsigned ukth = mi[0];                    // bits of the k-th largest (raw)

  // ---- collect survivors (raw >= kth), ~50 of them ----
  for (int j = tid * 4; j < VPAD_; j += NT_ * 4) {
    float4 v = *(const float4*)(store + j);
    float e[4] = {v.x, v.y, v.z, v.w};
#pragma unroll
    for (int t = 0; t < 4; ++t) {
      if (fmap(e[t]) >= ukth) {
        unsigned ix = atomicAdd(&mi[2], 1u);
        if (ix < SURV_CAP) surv[ix] = e[t];
      }
    }
  }
  __syncthreads();

  // ---- serial tail (n<=~50): sort desc, temperature, nucleus cut, logZ ----
  if (tid == 0) {
    int n = (int)mi[2]; if (n > SURV_CAP) n = SURV_CAP; if (n < 1) n = 1;
    for (int i = 1; i < n; ++i) {                 // insertion sort, descending
      float v = surv[i]; int j = i - 1;
      while (j >= 0 && surv[j] < v) { surv[j + 1] = surv[j]; --j; }
      surv[j + 1] = v;
    }
    float M = surv[0] / TEMP_;                    // scaled max
    float Zall = 0.f;
    for (int i = 0; i < n; ++i) Zall += expf(surv[i] / TEMP_ - M);
    float th = TOPP_ * Zall;                      // cum_prob > 0.9  <=>  cum > 0.9*Z
    int kept = n; float cum = 0.f;
    for (int j = 1; j < n; ++j) {                 // remove[j] <=> cum(<=j-1) > th
      cum += expf(surv[j - 1] / TEMP_ - M);
      if (cum > th) { kept = j; break; }
    }
    float Zk = 0.f;
    for (int i = 0; i < kept; ++i) Zk += expf(surv[i] / TEMP_ - M);
    mf[0] = surv[kept - 1];                       // raw-domain keep threshold
    mf[1] = M + logf(Zk);                         // logZ of final scaled row
  }
  __syncthreads();

  // ---- output: lazy temperature + threshold + log-softmax, b128 stores ----
  const float T2raw = mf[0], lZ = mf[1];
  const size_t base = (size_t)r * (size_t)V_;
  if (tid < a0) {                                 // lead (0..3 elems)
    float yr = row[tid];
    out[base + tid] = (yr >= T2raw) ? (yr / TEMP_ - lZ) : (FILT_ - lZ);
  }
  for (int q = tid; q < nq; q += NT_) {           // aligned body
    int i = a0 + 4 * q;
    float4 v = *(const float4*)(row + i);         // LDS-aligned (sh+a0 ≡ 0 mod 4)
    float4 o;
    o.x = (v.x >= T2raw) ? (v.x / TEMP_ - lZ) : (FILT_ - lZ);
    o.y = (v.y >= T2raw) ? (v.y / TEMP_ - lZ) : (FILT_ - lZ);
    o.z = (v.z >= T2raw) ? (v.z / TEMP_ - lZ) : (FILT_ - lZ);
    o.w = (v.w >= T2raw) ? (v.w / TEMP_ - lZ) : (FILT_ - lZ);
    *(float4*)(out + base + i) = o;               // global-16B-aligned b128 store
  }
  if (tid < V_ - t0) {                            // tail (0..3 elems)
    float yr = row[t0 + tid];
    out[base + t0 + tid] = (yr >= T2raw) ? (yr / TEMP_ - lZ) : (FILT_ - lZ);
  }
}

// ---------------- host launch ----------------
extern "C" void kernel_launch(void* const* d_in, const int* in_sizes, int n_in,
                              void* d_out, int out_size, void* d_ws, size_t ws_size,
                              hipStream_t stream) {
  (void)in_sizes; (void)n_in; (void)out_size; (void)ws_size;
  const float* logits = (const float*)d_in[0];
  const int*   prev   = (const int*)d_in[1];
  float*       out    = (float*)d_out;
  float*       pen    = (float*)d_ws;   // B*S*P floats = 256 KB scratch

  (void)hipFuncSetAttribute((const void*)row_kernel,
                            hipFuncAttributeMaxDynamicSharedMemorySize, SMEMB_);

  pen_kernel<<<dim3((S_ * P_ + NT_ - 1) / NT_), dim3(NT_), 0, stream>>>(logits, prev, pen);
  row_kernel<<<dim3(B_ * S_), dim3(NT_), SMEMB_, stream>>>(logits, prev, pen, out);
}